// DynamicGraphConv_5909875000349
// MI455X (gfx1250) — compile-verified
//
#include <hip/hip_runtime.h>
#include <hip/hip_bf16.h>

typedef float v2f __attribute__((ext_vector_type(2)));
typedef float v8f __attribute__((ext_vector_type(8)));

namespace {
constexpr int kB = 4, kN = 2000, kL = 32, kD = 64;
constexpr int kH = 4, kND = 64, kK = 20, kDOUT = 64;
constexpr float kEPS = 1e-8f;
constexpr int kNT = kN / 16;        // 125 tiles per N dimension
constexpr int kBL = kB * kL;        // 128
}

// ---------------------------------------------------------------------------
// Kernel 1: S[h,i,j] = relu(e1[h,i,:] . e2[h,:,j]) / temperature[h]
// One wave32 per 16x16 output tile; fp32 WMMA, K-loop 64 in steps of 4.
// ---------------------------------------------------------------------------
__global__ __launch_bounds__(32) void scores_wmma_kernel(
    const float* __restrict__ e1, const float* __restrict__ e2,
    const float* __restrict__ temperature, float* __restrict__ S) {
  const int ti = blockIdx.x, tj = blockIdx.y, h = blockIdx.z;
  const int lane  = threadIdx.x;
  const int m     = lane & 15;
  const int khalf = lane >> 4;             // 0: K{0,1}, 1: K{2,3}
  const int i0 = ti * 16, j0 = tj * 16;
  const float* A  = e1 + (size_t)h * kN * kND;   // [N][ND]
  const float* Bm = e2 + (size_t)h * kND * kN;   // [ND][N]
  v8f c = {};
  for (int k = 0; k < kND; k += 4) {
    const int ka = k + khalf * 2;
    v2f a, b;
    a.x = A[(i0 + m) * kND + ka];
    a.y = A[(i0 + m) * kND + ka + 1];
    b.x = Bm[(size_t)ka * kN + (j0 + m)];
    b.y = Bm[(size_t)(ka + 1) * kN + (j0 + m)];
    c = __builtin_amdgcn_wmma_f32_16x16x4_f32(false, a, false, b,
                                              (short)0, c, false, false);
  }
  const float invt = 1.0f / temperature[h];
  float* out = S + ((size_t)h * kN + i0) * kN + j0;
  for (int r = 0; r < 8; ++r) {
    const int row = r + khalf * 8;         // C/D layout: M = vgpr + 8*half
    float v = c[r];
    v = v > 0.0f ? v * invt : 0.0f;
    out[(size_t)row * kN + m] = v;
  }
}

// ---------------------------------------------------------------------------
// Kernel 2: per (h,row): softmax over N, top-K mask, renormalize (in place).
// Row cached in LDS; K argmax rounds with negation-marking, ties -> low index.
// ---------------------------------------------------------------------------
__global__ __launch_bounds__(256) void softmax_topk_kernel(float* __restrict__ S) {
  const int row = blockIdx.x, h = blockIdx.y;
  float* srow = S + ((size_t)h * kN + row) * kN;
  const int t = threadIdx.x;

  __shared__ float prow[kN];
  __shared__ float red[256];
  __shared__ int   redi[256];

  for (int j = t; j < kN; j += 256) prow[j] = srow[j];
  __syncthreads();

  // row max
  float mx = -1e30f;
  for (int j = t; j < kN; j += 256) mx = fmaxf(mx, prow[j]);
  red[t] = mx; __syncthreads();
  for (int s = 128; s > 0; s >>= 1) {
    if (t < s) red[t] = fmaxf(red[t], red[t + s]);
    __syncthreads();
  }
  mx = red[0]; __syncthreads();

  // exp + sum (keep p in LDS)
  float sum = 0.0f;
  for (int j = t; j < kN; j += 256) {
    const float p = __expf(prow[j] - mx);
    prow[j] = p;                           // p > 0 always
    sum += p;
  }
  red[t] = sum; __syncthreads();
  for (int s = 128; s > 0; s >>= 1) {
    if (t < s) red[t] += red[t + s];
    __syncthreads();
  }
  const float P = red[0]; __syncthreads();

  // top-K: K rounds of argmax (selected entries negated -> excluded)
  for (int r = 0; r < kK; ++r) {
    float bv = -1.0f; int bi = kN;
    for (int j = t; j < kN; j += 256) {
      const float v = prow[j];
      if (v > bv || (v == bv && j < bi)) { bv = v; bi = j; }
    }
    red[t] = bv; redi[t] = bi; __syncthreads();
    for (int s = 128; s > 0; s >>= 1) {
      if (t < s) {
        const float ov = red[t + s]; const int oi = redi[t + s];
        if (ov > red[t] || (ov == red[t] && oi < redi[t])) {
          red[t] = ov; redi[t] = oi;
        }
      }
      __syncthreads();
    }
    if (t == 0) prow[redi[0]] = -prow[redi[0]];
    __syncthreads();
  }

  // selected sum (normalized), then renormalize with EPS
  float ssum = 0.0f;
  for (int j = t; j < kN; j += 256) {
    const float v = prow[j];
    if (v < 0.0f) ssum += (-v) / P;
  }
  red[t] = ssum; __syncthreads();
  for (int s = 128; s > 0; s >>= 1) {
    if (t < s) red[t] += red[t + s];
    __syncthreads();
  }
  const float denom = red[0] + kEPS; __syncthreads();

  for (int j = t; j < kN; j += 256) {
    const float v = prow[j];
    srow[j] = (v < 0.0f) ? ((-v) / P) / denom : 0.0f;
  }
}

// ---------------------------------------------------------------------------
// Kernel 3: per edge (i,j): tiny MLP over heads + sigmoid * mean(adj over h)
// ---------------------------------------------------------------------------
__global__ __launch_bounds__(256) void edge_mlp_kernel(
    const float* __restrict__ S, const float* __restrict__ ew1,
    const float* __restrict__ eb1, const float* __restrict__ ew2,
    const float* __restrict__ eb2, float* __restrict__ final_adj) {
  const size_t idx = (size_t)blockIdx.x * 256 + threadIdx.x;
  const size_t NN = (size_t)kN * kN;
  if (idx >= NN) return;
  const float f0 = S[idx];
  const float f1 = S[NN + idx];
  const float f2 = S[2 * NN + idx];
  const float f3 = S[3 * NN + idx];
  const float mean = 0.25f * (f0 + f1 + f2 + f3);
  float ew = eb2[0];
  for (int o = 0; o < 2 * kH; ++o) {
    float hid = eb1[o] + f0 * ew1[0 * 2 * kH + o] + f1 * ew1[1 * 2 * kH + o] +
                f2 * ew1[2 * 2 * kH + o] + f3 * ew1[3 * 2 * kH + o];
    hid = fmaxf(hid, 0.0f);
    ew += hid * ew2[o];
  }
  final_adj[idx] = mean / (1.0f + __expf(-ew));
}

// ---------------------------------------------------------------------------
// Kernel 3b: faT[i,j] = final_adj[j,i]  (LDS-tiled, coalesced both ways).
// Makes the agg A-fragment loads contiguous (global_load_b64).
// ---------------------------------------------------------------------------
__global__ __launch_bounds__(256) void transpose_kernel(
    const float* __restrict__ src, float* __restrict__ dst) {
  __shared__ float tile[16][17];           // +1 pad: no LDS bank conflicts
  const int tx = threadIdx.x & 15, ty = threadIdx.x >> 4;
  const int bx = blockIdx.x * 16, by = blockIdx.y * 16;
  tile[ty][tx] = src[(size_t)(by + ty) * kN + (bx + tx)];
  __syncthreads();
  dst[(size_t)(bx + ty) * kN + (by + tx)] = tile[tx][ty];
}

// ---------------------------------------------------------------------------
// Kernel 4: support[bl, n, :] = x[b, n, l, :] @ weight   (fp32 WMMA)
// grid: (n-tile 125, dout-tile 4, bl 128); one wave per 16x16 tile.
// ---------------------------------------------------------------------------
__global__ __launch_bounds__(32) void support_wmma_kernel(
    const float* __restrict__ x, const float* __restrict__ weight,
    float* __restrict__ support) {
  const int tn = blockIdx.x, td = blockIdx.y, bl = blockIdx.z;
  const int lane  = threadIdx.x;
  const int m     = lane & 15;
  const int khalf = lane >> 4;
  const int n0 = tn * 16, d0 = td * 16;
  const int b = bl >> 5, l = bl & 31;      // kL == 32
  v8f c = {};
  for (int k = 0; k < kD; k += 4) {
    const int ka = k + khalf * 2;
    v2f a, bfrag;
    const size_t xrow = (((size_t)b * kN + (n0 + m)) * kL + l) * kD;
    a.x = x[xrow + ka];
    a.y = x[xrow + ka + 1];
    bfrag.x = weight[(size_t)ka * kDOUT + (d0 + m)];
    bfrag.y = weight[(size_t)(ka + 1) * kDOUT + (d0 + m)];
    c = __builtin_amdgcn_wmma_f32_16x16x4_f32(false, a, false, bfrag,
                                              (short)0, c, false, false);
  }
  for (int r = 0; r < 8; ++r) {
    const int row = r + khalf * 8;
    support[((size_t)bl * kN + (n0 + row)) * kDOUT + (d0 + m)] = c[r];
  }
}

// ---------------------------------------------------------------------------
// Kernel 5: out[b,i,l,:] = sum_j final_adj[j,i] * support[bl,j,:] + bias
// Each wave computes a 16(i) x 64(dout) tile with 4 accumulators:
//   per K-step: 1x b64 A-load (from faT, contiguous) + 8x b32 B-loads
//   feeding 4 independent WMMAs (no D->A/B hazards between them).
// Operands are L2-resident (192MB L2 >> 16MB adj + 65MB support).
// ---------------------------------------------------------------------------
__global__ __launch_bounds__(32) void agg_wmma_kernel(
    const float* __restrict__ faT, const float* __restrict__ support,
    const float* __restrict__ bias, float* __restrict__ out) {
  const int ti = blockIdx.x, bl = blockIdx.y;
  const int lane  = threadIdx.x;
  const int m     = lane & 15;
  const int khalf = lane >> 4;
  const int i0 = ti * 16;
  const int b = bl >> 5, l = bl & 31;
  const float* arow = faT + (size_t)(i0 + m) * kN;        // A[m][k], contiguous k
  const float* srow = support + (size_t)bl * kN * kDOUT;  // B[k][dout]
  v8f c0 = {}, c1 = {}, c2 = {}, c3 = {};
  for (int j = 0; j < kN; j += 4) {
    const int ka = j + khalf * 2;
    v2f a;
    a.x = arow[ka];
    a.y = arow[ka + 1];                                   // -> global_load_b64
    const float* s0 = srow + (size_t)ka * kDOUT + m;
    const float* s1 = s0 + kDOUT;
    v2f b0, b1, b2, b3;
    b0.x = s0[0];  b0.y = s1[0];
    b1.x = s0[16]; b1.y = s1[16];
    b2.x = s0[32]; b2.y = s1[32];
    b3.x = s0[48]; b3.y = s1[48];
    c0 = __builtin_amdgcn_wmma_f32_16x16x4_f32(false, a, false, b0, (short)0, c0, false, false);
    c1 = __builtin_amdgcn_wmma_f32_16x16x4_f32(false, a, false, b1, (short)0, c1, false, false);
    c2 = __builtin_amdgcn_wmma_f32_16x16x4_f32(false, a, false, b2, (short)0, c2, false, false);
    c3 = __builtin_amdgcn_wmma_f32_16x16x4_f32(false, a, false, b3, (short)0, c3, false, false);
  }
  const float bi0 = bias[m], bi1 = bias[16 + m], bi2 = bias[32 + m], bi3 = bias[48 + m];
  for (int r = 0; r < 8; ++r) {
    const int i = i0 + r + khalf * 8;
    float* orow = out + (((size_t)b * kN + i) * kL + l) * kDOUT;
    orow[m]      = c0[r] + bi0;
    orow[16 + m] = c1[r] + bi1;
    orow[32 + m] = c2[r] + bi2;
    orow[48 + m] = c3[r] + bi3;
  }
}

// ---------------------------------------------------------------------------
extern "C" void kernel_launch(void* const* d_in, const int* in_sizes, int n_in,
                              void* d_out, int out_size, void* d_ws, size_t ws_size,
                              hipStream_t stream) {
  const float* x           = (const float*)d_in[0];
  const float* e1          = (const float*)d_in[1];
  const float* e2          = (const float*)d_in[2];
  const float* temperature = (const float*)d_in[3];
  const float* ew1         = (const float*)d_in[4];
  const float* eb1         = (const float*)d_in[5];
  const float* ew2         = (const float*)d_in[6];
  const float* eb2         = (const float*)d_in[7];
  const float* weight      = (const float*)d_in[8];
  const float* bias        = (const float*)d_in[9];

  float* out       = (float*)d_out;                                  // (B,N,L,DOUT)
  float* final_adj = out + (size_t)kB * kN * kL * kDOUT;             // (N,N)

  float* S       = (float*)d_ws;                                     // (H,N,N)   64 MB
  float* support = S + (size_t)kH * kN * kN;                         // (BL,N,DOUT) 65.5 MB
  float* faT     = support + (size_t)kBL * kN * kDOUT;               // (N,N)^T   16 MB

  scores_wmma_kernel<<<dim3(kNT, kNT, kH), 32, 0, stream>>>(e1, e2, temperature, S);
  softmax_topk_kernel<<<dim3(kN, kH), 256, 0, stream>>>(S);
  edge_mlp_kernel<<<(int)(((size_t)kN * kN + 255) / 256), 256, 0, stream>>>(
      S, ew1, eb1, ew2, eb2, final_adj);
  transpose_kernel<<<dim3(kNT, kNT), 256, 0, stream>>>(final_adj, faT);
  support_wmma_kernel<<<dim3(kNT, kDOUT / 16, kBL), 32, 0, stream>>>(x, weight, support);
  agg_wmma_kernel<<<dim3(kNT, kBL), 32, 0, stream>>>(faT, support, bias, out);
}